// SparseKHopGraphAttention_8143257994120
// MI455X (gfx1250) — compile-verified
//
#include <hip/hip_runtime.h>
#include <hip/hip_bf16.h>
#include <math.h>

typedef float v2f __attribute__((ext_vector_type(2)));
typedef float v8f __attribute__((ext_vector_type(8)));

#define NNODES 50000
#define NEDGES 800000
#define DMODEL 128
#define NHEAD  8
#define HDIM   16
#define DFFN   384

// ---------------------------------------------------------------------------
// f32 WMMA GEMM with LDS-staged weights:
//   Y[M,NOUT] = A[M,KTOT] @ W[KTOT,NOUT] + bias (opt. ReLU)
// Block = 256 threads = 8 waves -> 128 rows x 64 cols output tile.
// Weight panel staged in LDS in K-paired layout so a wave's B fragment is one
// ds_load_b64; A fragment is one global b64. Uses V_WMMA_F32_16X16X4_F32.
//   A frag (16x4): lane l holds M=l%16, K = 2*(l/16)+{0,1}
//   B frag (4x16): lane l holds N=l%16, K = 2*(l/16)+{0,1}
//   D      (16x16): lane l, vgpr v -> row = 8*(l/16)+v, col = l%16
// ---------------------------------------------------------------------------
template <int KTOT, int NOUT, bool RELU>
__global__ __launch_bounds__(256) void gemm_bias_wmma(
    const float* __restrict__ A, const float* __restrict__ W,
    const float* __restrict__ bias, float* __restrict__ Y, int M) {
  constexpr int KC = 128;                 // K chunk staged per barrier
  __shared__ float2 ldsB[(KC / 2) * 64];  // 32 KB: [k2][col] = {W[2k2][c], W[2k2+1][c]}

  constexpr int tilesN = NOUT / 64;
  const int bm    = blockIdx.x / tilesN;
  const int bn    = blockIdx.x % tilesN;
  const int wslot = threadIdx.x >> 5;
  const int lane  = threadIdx.x & 31;
  const int l16   = lane & 15;
  const int khalf = lane >> 4;  // 0 or 1
  const int colBase = bn * 64;

  // clamp A row so EXEC stays all-ones through WMMA and barriers
  const int rowA  = bm * 128 + wslot * 16 + l16;
  const int rowAc = rowA < M ? rowA : (M - 1);
  const float* aRow = A + (size_t)rowAc * KTOT;

  v8f acc[4];
#pragma unroll
  for (int j = 0; j < 4; ++j) acc[j] = (v8f){0.f, 0.f, 0.f, 0.f, 0.f, 0.f, 0.f, 0.f};

  for (int kc = 0; kc < KTOT; kc += KC) {
    __syncthreads();  // previous chunk fully consumed
    // cooperative fill: (KC/2)*64 = 4096 float2, 16 per thread, col-coalesced
#pragma unroll
    for (int i = threadIdx.x; i < (KC / 2) * 64; i += 256) {
      const int k2 = i >> 6, col = i & 63;
      const float* wp = W + (size_t)(kc + 2 * k2) * NOUT + colBase + col;
      float2 t;
      t.x = wp[0];
      t.y = wp[NOUT];
      ldsB[i] = t;
    }
    __syncthreads();

    if (kc + KC < KTOT)  // prefetch next A chunk (global_prefetch_b8)
      __builtin_prefetch(aRow + kc + KC, 0, 1);

#pragma unroll 8
    for (int k0 = 0; k0 < KC; k0 += 4) {
      v2f a = *(const v2f*)(aRow + kc + k0 + 2 * khalf);  // 8B aligned
      const int k2 = (k0 >> 1) + khalf;
#pragma unroll
      for (int j = 0; j < 4; ++j) {
        v2f b = *(const v2f*)&ldsB[k2 * 64 + j * 16 + l16];  // ds_load_b64
        acc[j] = __builtin_amdgcn_wmma_f32_16x16x4_f32(
            /*neg_a=*/false, a, /*neg_b=*/false, b,
            /*c_mod=*/(short)0, acc[j], /*reuse_a=*/false, /*reuse_b=*/false);
      }
    }
  }

  const int rbase = bm * 128 + wslot * 16 + 8 * khalf;
#pragma unroll
  for (int j = 0; j < 4; ++j) {
    const int col = colBase + j * 16 + l16;
    const float bv = bias[col];
#pragma unroll
    for (int v = 0; v < 8; ++v) {
      const int rr = rbase + v;
      float val = acc[j][v] + bv;
      if (RELU) val = fmaxf(val, 0.f);
      if (rr < M) Y[(size_t)rr * NOUT + col] = val;
    }
  }
}

// ---------------------------------------------------------------------------
// Init: smax = -inf, den = 0 (N*H each), agg = 0 (N*D)
// ---------------------------------------------------------------------------
__global__ void init_buffers(float* __restrict__ smax, float* __restrict__ den,
                             float* __restrict__ agg, int nh, int nd) {
  int t = blockIdx.x * blockDim.x + threadIdx.x;
  if (t < nh) {
    smax[t] = __int_as_float(0xff800000);  // -inf
    den[t]  = 0.f;
  }
  if (t < nd) agg[t] = 0.f;
}

// ---------------------------------------------------------------------------
// SDDMM: s[e,h] = <q[r[e],h,:], k[c[e],h,:]> / sqrt(16)
// ---------------------------------------------------------------------------
__global__ void sddmm_kernel(const float* __restrict__ q, const float* __restrict__ k,
                             const int* __restrict__ r, const int* __restrict__ c,
                             float* __restrict__ s, int EH) {
  int t = blockIdx.x * blockDim.x + threadIdx.x;
  if (t >= EH) return;
  const int e = t >> 3, h = t & 7;
  const float* qp = q + (size_t)r[e] * DMODEL + h * HDIM;
  const float* kp = k + (size_t)c[e] * DMODEL + h * HDIM;
  float acc = 0.f;
#pragma unroll
  for (int i = 0; i < HDIM; i += 4) {
    float4 a = *(const float4*)(qp + i);
    float4 b = *(const float4*)(kp + i);
    acc += a.x * b.x + a.y * b.y + a.z * b.z + a.w * b.w;
  }
  s[t] = acc * 0.25f;  // 1/sqrt(HDIM)
}

__device__ inline void atomicMaxF(float* addr, float value) {
  if (value >= 0.f)
    atomicMax((int*)addr, __float_as_int(value));
  else
    atomicMin((unsigned int*)addr, __float_as_uint(value));
}

__global__ void seg_max_kernel(const float* __restrict__ s, const int* __restrict__ r,
                               float* __restrict__ smax, int EH) {
  int t = blockIdx.x * blockDim.x + threadIdx.x;
  if (t >= EH) return;
  const int e = t >> 3, h = t & 7;
  atomicMaxF(&smax[r[e] * NHEAD + h], s[t]);
}

// p = exp(s - smax[r]) written in place over s; den += p
__global__ void exp_den_kernel(float* __restrict__ s, const float* __restrict__ smax,
                               const int* __restrict__ r, float* __restrict__ den, int EH) {
  int t = blockIdx.x * blockDim.x + threadIdx.x;
  if (t >= EH) return;
  const int e = t >> 3, h = t & 7;
  const float p = expf(s[t] - smax[r[e] * NHEAD + h]);
  s[t] = p;
  atomicAdd(&den[r[e] * NHEAD + h], p);
}

// agg[r[e], h, :] += (p[e,h]/den[r[e],h]) * v[c[e], h, :]
__global__ void attn_scatter_kernel(const float* __restrict__ p, const float* __restrict__ den,
                                    const float* __restrict__ v, const int* __restrict__ r,
                                    const int* __restrict__ c, float* __restrict__ agg, int EH) {
  int t = blockIdx.x * blockDim.x + threadIdx.x;
  if (t >= EH) return;
  const int e = t >> 3, h = t & 7;
  const int rr = r[e], cc = c[e];
  const float w = p[t] / den[rr * NHEAD + h];
  const float* vp = v + (size_t)cc * DMODEL + h * HDIM;
  float* ap = agg + (size_t)rr * DMODEL + h * HDIM;
#pragma unroll
  for (int i = 0; i < HDIM; i += 4) {
    float4 vv = *(const float4*)(vp + i);
    atomicAdd(ap + i + 0, w * vv.x);
    atomicAdd(ap + i + 1, w * vv.y);
    atomicAdd(ap + i + 2, w * vv.z);
    atomicAdd(ap + i + 3, w * vv.w);
  }
}

// ---------------------------------------------------------------------------
// LayerNorm over rows of width 128: one wave32 per row, float4 per lane.
// ---------------------------------------------------------------------------
__global__ __launch_bounds__(256) void layernorm128(const float* __restrict__ X,
                                                    const float* __restrict__ g,
                                                    const float* __restrict__ b,
                                                    float* __restrict__ Y, int rows) {
  const int wv   = (blockIdx.x * blockDim.x + threadIdx.x) >> 5;
  const int lane = threadIdx.x & 31;
  if (wv >= rows) return;  // wave-uniform
  float4 xv = ((const float4*)(X + (size_t)wv * DMODEL))[lane];
  float s = xv.x + xv.y + xv.z + xv.w;
#pragma unroll
  for (int o = 16; o > 0; o >>= 1) s += __shfl_xor(s, o, 32);
  const float mean = s * (1.f / 128.f);
  const float d0 = xv.x - mean, d1 = xv.y - mean, d2 = xv.z - mean, d3 = xv.w - mean;
  float vs = d0 * d0 + d1 * d1 + d2 * d2 + d3 * d3;
#pragma unroll
  for (int o = 16; o > 0; o >>= 1) vs += __shfl_xor(vs, o, 32);
  const float inv = rsqrtf(vs * (1.f / 128.f) + 1e-5f);
  float4 gv = ((const float4*)g)[lane];
  float4 bv = ((const float4*)b)[lane];
  float4 out;
  out.x = d0 * inv * gv.x + bv.x;
  out.y = d1 * inv * gv.y + bv.y;
  out.z = d2 * inv * gv.z + bv.z;
  out.w = d3 * inv * gv.w + bv.w;
  ((float4*)(Y + (size_t)wv * DMODEL))[lane] = out;
}

// ---------------------------------------------------------------------------
static inline int cdiv(long long a, long long b) { return (int)((a + b - 1) / b); }

extern "C" void kernel_launch(void* const* d_in, const int* in_sizes, int n_in,
                              void* d_out, int out_size, void* d_ws, size_t ws_size,
                              hipStream_t stream) {
  const float* x   = (const float*)d_in[0];
  const int*   ei  = (const int*)d_in[1];
  const float* Wq  = (const float*)d_in[2];
  const float* bq  = (const float*)d_in[3];
  const float* Wk  = (const float*)d_in[4];
  const float* bk  = (const float*)d_in[5];
  const float* Wv  = (const float*)d_in[6];
  const float* bv  = (const float*)d_in[7];
  const float* W1  = (const float*)d_in[8];
  const float* b1  = (const float*)d_in[9];
  const float* W2  = (const float*)d_in[10];
  const float* b2  = (const float*)d_in[11];
  const float* g1  = (const float*)d_in[12];
  const float* be1 = (const float*)d_in[13];
  const float* g2  = (const float*)d_in[14];
  const float* be2 = (const float*)d_in[15];

  const int* r = ei;            // edge_index[0]
  const int* c = ei + NEDGES;   // edge_index[1]

  const size_t ND = (size_t)NNODES * DMODEL;   // 6.4M
  const size_t EH = (size_t)NEDGES * NHEAD;    // 6.4M
  const size_t NH = (size_t)NNODES * NHEAD;    // 0.4M

  float* ws   = (float*)d_ws;
  float* q    = ws;            // ND
  float* k    = q + ND;        // ND
  float* v    = k + ND;        // ND
  float* s    = v + ND;        // EH (becomes p in place)
  float* smax = s + EH;        // NH
  float* den  = smax + NH;     // NH
  float* agg  = den + NH;      // ND
  float* h1   = agg + ND;      // NNODES*DFFN
  float* ln1  = q;             // reuse q after SDDMM consumed it
  float* h2   = k;             // reuse k

  const int EHi   = (int)EH;
  const int bmT   = cdiv(NNODES, 128);           // 391 row-blocks
  const int lnBlk = cdiv((long long)NNODES * 32, 256);

  // QKV projections (f32 WMMA, LDS-staged weights)
  gemm_bias_wmma<DMODEL, DMODEL, false><<<bmT * (DMODEL / 64), 256, 0, stream>>>(x, Wq, bq, q, NNODES);
  gemm_bias_wmma<DMODEL, DMODEL, false><<<bmT * (DMODEL / 64), 256, 0, stream>>>(x, Wk, bk, k, NNODES);
  gemm_bias_wmma<DMODEL, DMODEL, false><<<bmT * (DMODEL / 64), 256, 0, stream>>>(x, Wv, bv, v, NNODES);

  // scatter-softmax buffers must be re-initialized every call
  init_buffers<<<cdiv((long long)ND, 256), 256, 0, stream>>>(smax, den, agg, (int)NH, (int)ND);

  sddmm_kernel<<<cdiv(EH, 256), 256, 0, stream>>>(q, k, r, c, s, EHi);
  seg_max_kernel<<<cdiv(EH, 256), 256, 0, stream>>>(s, r, smax, EHi);
  exp_den_kernel<<<cdiv(EH, 256), 256, 0, stream>>>(s, smax, r, den, EHi);
  attn_scatter_kernel<<<cdiv(EH, 256), 256, 0, stream>>>(s, den, v, r, c, agg, EHi);

  layernorm128<<<lnBlk, 256, 0, stream>>>(agg, g1, be1, ln1, NNODES);

  // FFN (f32 WMMA) + final LN into d_out
  gemm_bias_wmma<DMODEL, DFFN, true><<<bmT * (DFFN / 64), 256, 0, stream>>>(ln1, W1, b1, h1, NNODES);
  gemm_bias_wmma<DFFN, DMODEL, false><<<bmT * (DMODEL / 64), 256, 0, stream>>>(h1, W2, b2, h2, NNODES);
  layernorm128<<<lnBlk, 256, 0, stream>>>(h2, g2, be2, (float*)d_out, NNODES);
}